// GCN_22084721836889
// MI455X (gfx1250) — compile-verified
//
#include <hip/hip_runtime.h>
#include <hip/hip_bf16.h>

typedef __attribute__((ext_vector_type(2))) float v2f;
typedef __attribute__((ext_vector_type(8))) float v8f;

#define DD 128
#define EPSV 1e-5f

__device__ __forceinline__ void atomAddF(float* p, float v) {
  unsafeAtomicAdd(p, v);   // hardware global_atomic_add_f32
}

// ---------------- utility kernels ----------------

__global__ __launch_bounds__(256) void k_fill(float* __restrict__ p, int n, float v) {
  int i = blockIdx.x * blockDim.x + threadIdx.x;
  if (i < n) p[i] = v;
}

__global__ __launch_bounds__(256) void k_init_h(const int* __restrict__ nfeat,
                                                const float* __restrict__ aemb,
                                                float* __restrict__ h, int n) {
  int i = blockIdx.x * blockDim.x + threadIdx.x;
  if (i >= n) return;
  int row = i >> 7, col = i & 127;
  h[i] = aemb[(size_t)nfeat[row] * DD + col];
}

__global__ __launch_bounds__(256) void k_count(const int* __restrict__ idx,
                                               float* __restrict__ acc, int n) {
  int i = blockIdx.x * blockDim.x + threadIdx.x;
  if (i < n) atomAddF(&acc[idx[i]], 1.0f);
}

__global__ __launch_bounds__(256) void k_recip_clamp(float* __restrict__ p, int n) {
  int i = blockIdx.x * blockDim.x + threadIdx.x;
  if (i < n) p[i] = 1.0f / fmaxf(p[i], 1.0f);
}

// pre-pack a 128x128 row-major W into B-fragment order:
// Wp[((kk*8 + t)*32 + lane)] = { W[kb+2h][t*16+n], W[kb+2h+1][t*16+n] } (float2)
// where kb = 4*kk, n = lane&15, h = lane>>4.
__global__ __launch_bounds__(256) void k_packW(const float* __restrict__ W,
                                               float* __restrict__ Wp) {
  int i = blockIdx.x * blockDim.x + threadIdx.x;   // 32*8*32 = 8192
  if (i >= 8192) return;
  int lane = i & 31;
  int t = (i >> 5) & 7;
  int kk = i >> 8;
  int n = lane & 15, hh = lane >> 4;
  int kb = kk * 4;
  Wp[2 * i + 0] = W[(size_t)(kb + 2 * hh + 0) * DD + t * 16 + n];
  Wp[2 * i + 1] = W[(size_t)(kb + 2 * hh + 1) * DD + t * 16 + n];
}

// scatter: neigh[dst] += h[src] + edge_emb[efeat]   (thread = edge x 4 cols)
__global__ __launch_bounds__(256) void k_scatter(const float* __restrict__ h,
                                                 const float* __restrict__ eemb,
                                                 const int* __restrict__ src,
                                                 const int* __restrict__ dst,
                                                 const int* __restrict__ efeat,
                                                 float* __restrict__ neigh, int nE) {
  int tid = blockIdx.x * blockDim.x + threadIdx.x;
  if (tid >= nE * 32) return;
  int e = tid >> 5;
  int c4 = (tid & 31) << 2;
  int s = src[e], d = dst[e], f = efeat[e];
  const float4 hv = *(const float4*)(h + (size_t)s * DD + c4);
  const float4 ev = *(const float4*)(eemb + (size_t)f * DD + c4);
  float* np = neigh + (size_t)d * DD + c4;
  atomAddF(np + 0, hv.x + ev.x);
  atomAddF(np + 1, hv.y + ev.y);
  atomAddF(np + 2, hv.z + ev.z);
  atomAddF(np + 3, hv.w + ev.w);
}

// pooling: pool[n2g[i]] += h[i]
__global__ __launch_bounds__(256) void k_poolsum(const float* __restrict__ h,
                                                 const int* __restrict__ n2g,
                                                 float* __restrict__ pool, int nN) {
  int tid = blockIdx.x * blockDim.x + threadIdx.x;
  if (tid >= nN * 32) return;
  int i = tid >> 5;
  int c4 = (tid & 31) << 2;
  int g = n2g[i];
  const float4 hv = *(const float4*)(h + (size_t)i * DD + c4);
  float* pp = pool + (size_t)g * DD + c4;
  atomAddF(pp + 0, hv.x);
  atomAddF(pp + 1, hv.y);
  atomAddF(pp + 2, hv.z);
  atomAddF(pp + 3, hv.w);
}

// batch-norm column stats: thread t owns column t, register partial sums
__global__ __launch_bounds__(128) void k_bnstats(const float* __restrict__ y,
                                                 float* __restrict__ sums, int nrows) {
  int col = threadIdx.x;
  float s = 0.f, s2 = 0.f;
  for (int r = blockIdx.x; r < nrows; r += gridDim.x) {
    float v = y[(size_t)r * DD + col];
    s += v;
    s2 += v * v;
  }
  atomAddF(&sums[col], s);
  atomAddF(&sums[DD + col], s2);
}

__global__ __launch_bounds__(256) void k_bnapply(const float* __restrict__ y,
                                                 const float* __restrict__ sums,
                                                 const float* __restrict__ gamma,
                                                 const float* __restrict__ beta,
                                                 float* __restrict__ h, int n, float invN) {
  int i = blockIdx.x * blockDim.x + threadIdx.x;
  if (i >= n) return;
  int col = i & 127;
  float mu = sums[col] * invN;
  float var = sums[DD + col] * invN - mu * mu;
  float v = (y[i] - mu) * rsqrtf(var + EPSV) * gamma[col] + beta[col];
  h[i] = fmaxf(v, 0.f);
}

// ---------------- WMMA GEMM: out[r][n] = act( x[r][:] @ W + bias[n] ) ----------------
// x[r][k] = (A0[r][k] + (A1?A1[r][k]:0)) * (rowscale?rowscale[r]:1)
// Wp is the fragment-packed 128x128 weight (see k_packW).
// K = Ncols = 128.  Block = 128 threads (4 waves); wave w owns rows
// [blk*64 + w*16, +16).  Uses V_WMMA_F32_16X16X4_F32, 32 k-steps, 8 n-tiles.
__global__ __launch_bounds__(128) void k_gemm128(const float* __restrict__ A0,
                                                 const float* __restrict__ A1,
                                                 const float* __restrict__ rowscale,
                                                 const float* __restrict__ Wp,
                                                 const float* __restrict__ bias,
                                                 float* __restrict__ out,
                                                 int nrows, int relu) {
  __shared__ float xs[4][16][132];   // +4 pad: conflict-free b64 A-frag reads
  const int lane = threadIdx.x & 31;
  const int wave = threadIdx.x >> 5;
  const int nm = lane & 15;          // M for A-frag / N for B-frag & D
  const int hh = lane >> 4;          // k-pair / row-half selector
  const int rowBase = blockIdx.x * 64 + wave * 16;

  // stage x tile (16 rows x 128) for this wave into LDS
  #pragma unroll 4
  for (int i = 0; i < 64; ++i) {
    int idx = lane + 32 * i;         // 0..2047
    int m = idx >> 7;
    int k = idx & 127;
    int row = rowBase + m;
    float v = 0.f;
    if (row < nrows) {
      v = A0[(size_t)row * DD + k];
      if (A1) v += A1[(size_t)row * DD + k];
      if (rowscale) v *= rowscale[row];
    }
    xs[wave][m][k] = v;
  }
  __syncthreads();

  v8f acc[8];
  #pragma unroll
  for (int t = 0; t < 8; ++t) acc[t] = (v8f){0.f, 0.f, 0.f, 0.f, 0.f, 0.f, 0.f, 0.f};

  #pragma unroll 4
  for (int kk = 0; kk < 32; ++kk) {
    const int kb = kk * 4;
    // A-frag: element (m,k): lane = m + 16*(k>>1), vgpr = k&1
    v2f a = *(const v2f*)&xs[wave][nm][kb + 2 * hh];
    const float* wrow = Wp + (size_t)kk * 512 + lane * 2;   // 8 tiles * 32 lanes * 2
    #pragma unroll
    for (int t = 0; t < 8; ++t) {
      v2f bf = *(const v2f*)(wrow + t * 64);                // coalesced b64 load
      acc[t] = __builtin_amdgcn_wmma_f32_16x16x4_f32(
          false, a, false, bf, (short)0, acc[t], false, false);
    }
  }

  // D layout: vgpr r -> row r + 8*hh, col = t*16 + nm
  #pragma unroll
  for (int t = 0; t < 8; ++t) {
    const int col = t * 16 + nm;
    const float bv = bias[col];
    #pragma unroll
    for (int r = 0; r < 8; ++r) {
      int row = rowBase + r + 8 * hh;
      if (row < nrows) {
        float v = acc[t][r] + bv;
        if (relu) v = fmaxf(v, 0.f);
        out[(size_t)row * DD + col] = v;
      }
    }
  }
}

// ---------------- host launcher ----------------

static inline unsigned nblk(long long n, int b) { return (unsigned)((n + b - 1) / b); }

extern "C" void kernel_launch(void* const* d_in, const int* in_sizes, int n_in,
                              void* d_out, int out_size, void* d_ws, size_t ws_size,
                              hipStream_t stream) {
  const int*   nfeat = (const int*)d_in[0];
  const int*   efeat = (const int*)d_in[1];
  const int*   src   = (const int*)d_in[2];
  const int*   dst   = (const int*)d_in[3];
  const int*   n2g   = (const int*)d_in[4];
  const float* aemb  = (const float*)d_in[5];
  const float* eemb  = (const float*)d_in[6];
  const float* W     = (const float*)d_in[7];
  const float* b     = (const float*)d_in[8];
  const float* gamma = (const float*)d_in[9];
  const float* beta  = (const float*)d_in[10];
  const float* W1    = (const float*)d_in[11];
  const float* b1    = (const float*)d_in[12];
  const float* W2    = (const float*)d_in[13];
  const float* b2    = (const float*)d_in[14];

  const int N  = in_sizes[0];
  const int E  = in_sizes[1];
  const int L  = in_sizes[7] / (DD * DD);
  const int BV = in_sizes[6] / (L * DD);
  const int G  = out_size / DD;
  const int WPSZ = 16384;   // floats per packed 128x128 weight

  // workspace layout (floats)
  float* h      = (float*)d_ws;
  float* neigh  = h + (size_t)N * DD;          // also GEMM output y (in-place)
  float* deg    = neigh + (size_t)N * DD;      // becomes invdeg
  float* bns    = deg + N;                     // 256: col sums / sumsq
  float* pool   = bns + 256;
  float* cnt    = pool + (size_t)G * DD;       // becomes 1/max(cnt,1)
  float* hidden = cnt + G;
  float* Wpk    = hidden + (size_t)G * DD;     // (L+2) packed weights

  const int TB = 256;
  const int ND = N * DD;

  // pack all weight matrices into B-fragment order
  for (int l = 0; l < L; ++l)
    k_packW<<<32, TB, 0, stream>>>(W + (size_t)l * DD * DD, Wpk + (size_t)l * WPSZ);
  k_packW<<<32, TB, 0, stream>>>(W1, Wpk + (size_t)L * WPSZ);
  k_packW<<<32, TB, 0, stream>>>(W2, Wpk + (size_t)(L + 1) * WPSZ);

  // init: h = atom_emb[nfeat]; invdeg = 1/(indeg+1)
  k_init_h<<<nblk(ND, TB), TB, 0, stream>>>(nfeat, aemb, h, ND);
  k_fill<<<nblk(N, TB), TB, 0, stream>>>(deg, N, 1.0f);
  k_count<<<nblk(E, TB), TB, 0, stream>>>(dst, deg, E);
  k_recip_clamp<<<nblk(N, TB), TB, 0, stream>>>(deg, N);

  const float invN = 1.0f / (float)N;
  for (int l = 0; l < L; ++l) {
    k_fill<<<nblk(ND, TB), TB, 0, stream>>>(neigh, ND, 0.0f);
    k_fill<<<1, 256, 0, stream>>>(bns, 256, 0.0f);
    k_scatter<<<nblk((long long)E * 32, TB), TB, 0, stream>>>(
        h, eemb + (size_t)l * BV * DD, src, dst, efeat, neigh, E);
    // y = ((h + neigh) * invdeg) @ W[l] + b[l]   (written in place over neigh)
    k_gemm128<<<nblk(N, 64), 128, 0, stream>>>(
        h, neigh, deg, Wpk + (size_t)l * WPSZ, b + (size_t)l * DD, neigh, N, 0);
    k_bnstats<<<512, DD, 0, stream>>>(neigh, bns, N);
    k_bnapply<<<nblk(ND, TB), TB, 0, stream>>>(
        neigh, bns, gamma + (size_t)l * DD, beta + (size_t)l * DD, h, ND, invN);
  }

  // global mean pool per graph
  k_fill<<<nblk(G * DD, TB), TB, 0, stream>>>(pool, G * DD, 0.0f);
  k_fill<<<nblk(G, TB), TB, 0, stream>>>(cnt, G, 0.0f);
  k_count<<<nblk(N, TB), TB, 0, stream>>>(n2g, cnt, N);
  k_poolsum<<<nblk((long long)N * 32, TB), TB, 0, stream>>>(h, n2g, pool, N);
  k_recip_clamp<<<nblk(G, TB), TB, 0, stream>>>(cnt, G);

  // MLP head: relu(g @ W1 + b1) @ W2 + b2
  k_gemm128<<<nblk(G, 64), 128, 0, stream>>>(pool, nullptr, cnt,
                                             Wpk + (size_t)L * WPSZ, b1, hidden, G, 1);
  k_gemm128<<<nblk(G, 64), 128, 0, stream>>>(hidden, nullptr, nullptr,
                                             Wpk + (size_t)(L + 1) * WPSZ, b2,
                                             (float*)d_out, G, 0);
}